// FrequencySpatialAdaptiveAttention_25778393710997
// MI455X (gfx1250) — compile-verified
//
#include <hip/hip_runtime.h>

// ---------------------------------------------------------------------------
// FrequencySpatialAdaptiveAttention for MI455X (gfx1250), fp32 end-to-end.
// B=2, N=4096, C=256. Dense GEMMs via V_WMMA_F32_16X16X4_F32 with
// GLOBAL_LOAD_ASYNC_TO_LDS_B128 tile staging (ASYNCcnt) when available;
// Laplacian kept sparse (kNN + reciprocal edges, cap 64/row) -> gather-SpMM.
// ---------------------------------------------------------------------------

namespace {
constexpr int BB  = 2;
constexpr int NNp = 4096;
constexpr int CCc = 256;
constexpr int MM  = BB * NNp;          // 8192 rows
constexpr int KNNk = 16;
constexpr int CAP  = 64;               // symmetric neighbor-list capacity
constexpr float EPSc = 1e-5f;
}

typedef float v2f __attribute__((ext_vector_type(2)));
typedef float v8f __attribute__((ext_vector_type(8)));
typedef int   v4i __attribute__((ext_vector_type(4)));
typedef v4i __attribute__((address_space(1)))* as1_v4i_p;   // global
typedef v4i __attribute__((address_space(3)))* as3_v4i_p;   // LDS

#if defined(__has_builtin)
#if __has_builtin(__builtin_amdgcn_global_load_async_to_lds_b128) && \
    __has_builtin(__builtin_amdgcn_s_wait_asynccnt)
#define USE_ASYNC_LDS 1
#endif
#endif

__device__ __forceinline__ float wave_sum32(float v) {
#pragma unroll
  for (int m = 16; m > 0; m >>= 1) v += __shfl_xor(v, m, 32);
  return v;
}

// ---------------------------------------------------------------------------
// 1) kNN: per-row 16 nearest neighbors (excl. self), register insertion sort.
// ---------------------------------------------------------------------------
__global__ __launch_bounds__(256) void knn_kernel(const float* __restrict__ pts,
                                                  int* __restrict__ knn_idx) {
  int tid = blockIdx.x * blockDim.x + threadIdx.x;
  if (tid >= MM) return;
  int b = tid / NNp, i = tid - b * NNp;
  const float* pb = pts + (size_t)b * NNp * 3;
  float px = pb[i * 3 + 0], py = pb[i * 3 + 1], pz = pb[i * 3 + 2];
  float bd[KNNk]; int bix[KNNk];
#pragma unroll
  for (int k = 0; k < KNNk; ++k) { bd[k] = 3.4e38f; bix[k] = 0; }
  for (int j = 0; j < NNp; ++j) {
    if (j == i) continue;
    float dx = pb[j * 3 + 0] - px, dy = pb[j * 3 + 1] - py, dz = pb[j * 3 + 2] - pz;
    float d = dx * dx + dy * dy + dz * dz;
    if (d < bd[KNNk - 1]) {
      float cd = d; int ci = j;
#pragma unroll
      for (int k = 0; k < KNNk; ++k) {          // branch-free insertion
        bool sw = cd < bd[k];
        float td = bd[k]; int ti = bix[k];
        bd[k] = sw ? cd : td; bix[k] = sw ? ci : ti;
        cd = sw ? td : cd;    ci = sw ? ti : ci;
      }
    }
  }
#pragma unroll
  for (int k = 0; k < KNNk; ++k) knn_idx[(size_t)tid * KNNk + k] = bix[k];
}

// ---------------------------------------------------------------------------
// 2) Symmetric adjacency: init with kNN, then insert reciprocal edges.
// ---------------------------------------------------------------------------
__global__ __launch_bounds__(256) void adj_init_kernel(const int* __restrict__ knn_idx,
                                                       int* __restrict__ cnt,
                                                       int* __restrict__ nbrs) {
  int tid = blockIdx.x * blockDim.x + threadIdx.x;
  if (tid >= MM) return;
  cnt[tid] = KNNk;
#pragma unroll
  for (int k = 0; k < KNNk; ++k)
    nbrs[(size_t)tid * CAP + k] = knn_idx[(size_t)tid * KNNk + k];
}

__global__ __launch_bounds__(256) void adj_recip_kernel(const int* __restrict__ knn_idx,
                                                        int* __restrict__ cnt,
                                                        int* __restrict__ nbrs) {
  int tid = blockIdx.x * blockDim.x + threadIdx.x;
  if (tid >= MM * KNNk) return;
  int row = tid / KNNk, kk = tid - row * KNNk;
  int b = row / NNp, il = row - b * NNp;
  int j = knn_idx[(size_t)row * KNNk + kk];      // batch-local neighbor
  int jrow = b * NNp + j;
  const int* kj = knn_idx + (size_t)jrow * KNNk;
  bool found = false;
#pragma unroll
  for (int t = 0; t < KNNk; ++t) found = found || (kj[t] == il);
  if (!found) {
    int pos = atomicAdd(&cnt[jrow], 1);
    if (pos < CAP) nbrs[(size_t)jrow * CAP + pos] = il;
  }
}

__global__ __launch_bounds__(256) void deg_kernel(int* __restrict__ cnt,
                                                  float* __restrict__ dinv) {
  int tid = blockIdx.x * blockDim.x + threadIdx.x;
  if (tid >= MM) return;
  int d = cnt[tid]; d = d < CAP ? d : CAP;
  cnt[tid] = d;
  dinv[tid] = rsqrtf(fmaxf((float)d, 1e-12f));
}

// ---------------------------------------------------------------------------
// 3) Sparse Laplacian apply: y = alpha*(L x) + beta*z,
//    (L x)_i = x_i - dinv_i * sum_j dinv_j * x_j.   One block per row.
// ---------------------------------------------------------------------------
__global__ __launch_bounds__(256) void spmm_kernel(const float* __restrict__ x,
                                                   const float* __restrict__ z,
                                                   const int* __restrict__ nbrs,
                                                   const int* __restrict__ cnt,
                                                   const float* __restrict__ dinv,
                                                   float* __restrict__ y,
                                                   float alpha, float beta) {
  __shared__ int   sj[CAP];
  __shared__ float sd[CAP];
  int row = blockIdx.x;
  int b = row / NNp, base = b * NNp;
  int c = threadIdx.x;                              // 256 threads == CCc
  int n = cnt[row];
  if (c < n) {
    int j = nbrs[(size_t)row * CAP + c];
    sj[c] = j;
    sd[c] = dinv[base + j];
  }
  __syncthreads();
  float di = dinv[row];
  float acc = x[(size_t)row * CCc + c];
  for (int t = 0; t < n; ++t)
    acc -= di * sd[t] * x[(size_t)(base + sj[t]) * CCc + c];
  float r = alpha * acc;
  if (beta != 0.0f) r += beta * z[(size_t)row * CCc + c];
  y[(size_t)row * CCc + c] = r;
}

// ---------------------------------------------------------------------------
// 4) fp32 WMMA GEMM:  C = (accumulate ? C : bias) + A[M,K] @ W[K,N]
//    Block 256 thr (8 waves) -> 128x64 tile, K staged 32-wide in LDS.
//    A tile staged via GLOBAL_LOAD_ASYNC_TO_LDS_B128 (ASYNCcnt) when the
//    builtin exists; W tile stored K-pair-interleaved so each B fragment is
//    a single 8-byte ds_load_b64. 32x v_wmma_f32_16x16x4_f32 per K stage.
// ---------------------------------------------------------------------------
__global__ __launch_bounds__(256) void gemm_wmma_f32(const float* __restrict__ A,
                                                     const float* __restrict__ W,
                                                     const float* __restrict__ bias,
                                                     float* __restrict__ Cm,
                                                     int K, int lda, int ldw, int ldc,
                                                     int accumulate) {
  // As stride 36 floats: row starts 16B-aligned (async b128 dst), v2f reads 8B
  // aligned, 36*m mod 64 hits 16 distinct banks for the A-fragment gather.
  __shared__ float As[128][36];
  // BsP[k/2][n][k&1]: kpair stride 160 floats == 32 banks, so the two lane
  // halves (adjacent kpairs) use disjoint bank halves. B frag = one b64 load.
  __shared__ float BsP[16][80][2];
  const int t = threadIdx.x;
  const int wv = t >> 5, lane = t & 31;
  const int half = lane >> 4, l15 = lane & 15;
  const int m0 = blockIdx.x * 128, n0 = blockIdx.y * 64;

  v8f acc[4];
#pragma unroll
  for (int i = 0; i < 4; ++i) acc[i] = (v8f){0.f,0.f,0.f,0.f,0.f,0.f,0.f,0.f};

  for (int k0 = 0; k0 < K; k0 += 32) {
    // ---- stage A 128x32 ----
#if defined(USE_ASYNC_LDS)
#pragma unroll
    for (int it = 0; it < 4; ++it) {
      int lin = t + 256 * it;                 // 0..1023
      int row = lin >> 3, c4 = (lin & 7) << 2;
      __builtin_amdgcn_global_load_async_to_lds_b128(
          (as1_v4i_p)(unsigned long long)&A[(size_t)(m0 + row) * lda + k0 + c4],
          (as3_v4i_p)(void*)&As[row][c4],
          0, 0);
    }
#else
#pragma unroll
    for (int it = 0; it < 4; ++it) {
      int lin = t + 256 * it;                 // 0..1023
      int row = lin >> 3, c4 = (lin & 7) << 2;
      const float4 va = *(const float4*)&A[(size_t)(m0 + row) * lda + k0 + c4];
      As[row][c4 + 0] = va.x; As[row][c4 + 1] = va.y;
      As[row][c4 + 2] = va.z; As[row][c4 + 3] = va.w;
    }
#endif
    // ---- stage W 32x64, K-pair interleaved ----
#pragma unroll
    for (int it = 0; it < 2; ++it) {
      int lin = t + 256 * it;                 // 0..511
      int row = lin >> 4, c4 = (lin & 15) << 2;
      const float4 vb = *(const float4*)&W[(size_t)(k0 + row) * ldw + n0 + c4];
      int kp = row >> 1, par = row & 1;
      BsP[kp][c4 + 0][par] = vb.x; BsP[kp][c4 + 1][par] = vb.y;
      BsP[kp][c4 + 2][par] = vb.z; BsP[kp][c4 + 3][par] = vb.w;
    }
    if (k0 + 32 < K)  // pull next A tile toward L2 (global_prefetch_b8)
      __builtin_prefetch(&A[(size_t)(m0 + (t & 127)) * lda + k0 + 32], 0, 0);
#if defined(USE_ASYNC_LDS)
    __builtin_amdgcn_s_wait_asynccnt(0);
#endif
    __syncthreads();

    const int mrow = wv * 16;
#pragma unroll
    for (int kk = 0; kk < 32; kk += 4) {
      const int ksel = kk + 2 * half;         // even
      // A frag (16x4 f32): v0 = A[m][ksel], v1 = A[m][ksel+1]
      v2f a = *(const v2f*)&As[mrow + l15][ksel];
#pragma unroll
      for (int ns = 0; ns < 4; ++ns) {
        v2f bfrag = *(const v2f*)&BsP[ksel >> 1][ns * 16 + l15][0];
        acc[ns] = __builtin_amdgcn_wmma_f32_16x16x4_f32(
            false, a, false, bfrag, (short)0, acc[ns], false, false);
      }
    }
    __syncthreads();
  }

  // ---- epilogue: C layout VGPR r -> M = r + 8*half, lane%16 -> N ----
  const int row_base = m0 + wv * 16 + 8 * half;
#pragma unroll
  for (int ns = 0; ns < 4; ++ns) {
    int col = n0 + ns * 16 + l15;
#pragma unroll
    for (int r = 0; r < 8; ++r) {
      size_t o = (size_t)(row_base + r) * ldc + col;
      float base = accumulate ? Cm[o] : bias[col];
      Cm[o] = base + acc[ns][r];
    }
  }
}

// ---------------------------------------------------------------------------
// 5) LayerNorm (one wave per row of 256): out = LN(x)*g+b, optional relu,
//    optional residual: out = resid + gamma0 * LN(...)
// ---------------------------------------------------------------------------
__global__ __launch_bounds__(256) void ln_kernel(const float* __restrict__ x,
                                                 const float* __restrict__ g,
                                                 const float* __restrict__ bln,
                                                 const float* __restrict__ resid,
                                                 const float* __restrict__ gamma,
                                                 float* __restrict__ out,
                                                 int do_relu) {
  int wv = (blockIdx.x * (blockDim.x >> 5)) + (threadIdx.x >> 5);
  int lane = threadIdx.x & 31;
  if (wv >= MM) return;
  const float* xr = x + (size_t)wv * CCc;
  float v[8], s = 0.f, s2 = 0.f;
#pragma unroll
  for (int u = 0; u < 8; ++u) {
    int c = lane + 32 * u;
    v[u] = xr[c];
    s += v[u]; s2 += v[u] * v[u];
  }
  s = wave_sum32(s); s2 = wave_sum32(s2);
  float mu = s * (1.0f / CCc);
  float var = s2 * (1.0f / CCc) - mu * mu;
  float rs = rsqrtf(var + EPSc);
  float gm = (gamma != nullptr) ? gamma[0] : 0.f;
#pragma unroll
  for (int u = 0; u < 8; ++u) {
    int c = lane + 32 * u;
    float yv = (v[u] - mu) * rs * g[c] + bln[c];
    if (do_relu) yv = fmaxf(yv, 0.f);
    if (resid != nullptr) yv = resid[(size_t)wv * CCc + c] + gm * yv;
    out[(size_t)wv * CCc + c] = yv;
  }
}

// ---------------------------------------------------------------------------
// 6) gate = softmax(h @ Wg2 + bg2)  (one wave per row; 3 outputs)
// ---------------------------------------------------------------------------
__global__ __launch_bounds__(256) void gate_kernel(const float* __restrict__ h,
                                                   const float* __restrict__ Wg2,
                                                   const float* __restrict__ bg2,
                                                   float* __restrict__ gate) {
  int wv = (blockIdx.x * (blockDim.x >> 5)) + (threadIdx.x >> 5);
  int lane = threadIdx.x & 31;
  if (wv >= MM) return;
  float a0 = 0.f, a1 = 0.f, a2 = 0.f;
#pragma unroll
  for (int u = 0; u < 8; ++u) {
    int c = lane + 32 * u;
    float hv = h[(size_t)wv * CCc + c];
    a0 += hv * Wg2[c * 3 + 0];
    a1 += hv * Wg2[c * 3 + 1];
    a2 += hv * Wg2[c * 3 + 2];
  }
  a0 = wave_sum32(a0); a1 = wave_sum32(a1); a2 = wave_sum32(a2);
  if (lane == 0) {
    float z0 = a0 + bg2[0], z1 = a1 + bg2[1], z2 = a2 + bg2[2];
    float m = fmaxf(z0, fmaxf(z1, z2));
    float e0 = expf(z0 - m), e1 = expf(z1 - m), e2 = expf(z2 - m);
    float inv = 1.0f / (e0 + e1 + e2);
    gate[(size_t)wv * 3 + 0] = e0 * inv;
    gate[(size_t)wv * 3 + 1] = e1 * inv;
    gate[(size_t)wv * 3 + 2] = e2 * inv;
  }
}

// ---------------------------------------------------------------------------
// 7) f_fused = g0*f_sp + g1*f_low + g2*f_high (elementwise)
// ---------------------------------------------------------------------------
__global__ __launch_bounds__(256) void fuse_kernel(const float* __restrict__ gate,
                                                   const float* __restrict__ fsp,
                                                   const float* __restrict__ flow,
                                                   const float* __restrict__ fhigh,
                                                   float* __restrict__ out) {
  size_t tid = (size_t)blockIdx.x * blockDim.x + threadIdx.x;
  if (tid >= (size_t)MM * CCc) return;
  size_t row = tid >> 8;
  float g0 = gate[row * 3 + 0], g1 = gate[row * 3 + 1], g2 = gate[row * 3 + 2];
  out[tid] = g0 * fsp[tid] + g1 * flow[tid] + g2 * fhigh[tid];
}

// ---------------------------------------------------------------------------
// Host orchestration
// ---------------------------------------------------------------------------
extern "C" void kernel_launch(void* const* d_in, const int* in_sizes, int n_in,
                              void* d_out, int out_size, void* d_ws, size_t ws_size,
                              hipStream_t stream) {
  (void)in_sizes; (void)n_in; (void)out_size; (void)ws_size;
  const float* points = (const float*)d_in[0];
  const float* feats  = (const float*)d_in[1];
  const float* Wsp    = (const float*)d_in[2];
  const float* bsp    = (const float*)d_in[3];
  const float* th_l   = (const float*)d_in[4];
  const float* b_low  = (const float*)d_in[5];
  const float* th_h   = (const float*)d_in[6];
  const float* b_high = (const float*)d_in[7];
  const float* Wg1    = (const float*)d_in[8];
  const float* bg1    = (const float*)d_in[9];
  const float* glng   = (const float*)d_in[10];
  const float* glnb   = (const float*)d_in[11];
  const float* Wg2    = (const float*)d_in[12];
  const float* bg2    = (const float*)d_in[13];
  const float* Wo     = (const float*)d_in[14];
  const float* bo     = (const float*)d_in[15];
  const float* olng   = (const float*)d_in[16];
  const float* olnb   = (const float*)d_in[17];
  const float* gamma  = (const float*)d_in[18];
  float* out = (float*)d_out;

  const size_t MC = (size_t)MM * CCc;
  float* ws    = (float*)d_ws;
  float* fsp   = ws;              // f_spatial
  float* tx1   = fsp + MC;        // T1 = L feats      (later reused as f_fused)
  float* tx2   = tx1 + MC;        // T2 = 2 L T1 - x   (later reused as o_pre)
  float* flow  = tx2 + MC;
  float* fhigh = flow + MC;
  float* hbuf  = fhigh + MC;      // h_pre -> h (LN in place)
  int*   knn_i = (int*)(hbuf + MC);
  int*   cnt   = knn_i + (size_t)MM * KNNk;
  int*   nbrs  = cnt + MM;
  float* dinv  = (float*)(nbrs + (size_t)MM * CAP);
  float* gateb = dinv + MM;       // MM*3

  const int CC2 = CCc * CCc;      // 65536: weight-matrix stride

  // ---- graph construction ----
  knn_kernel<<<MM / 256, 256, 0, stream>>>(points, knn_i);
  adj_init_kernel<<<MM / 256, 256, 0, stream>>>(knn_i, cnt, nbrs);
  adj_recip_kernel<<<MM * KNNk / 256, 256, 0, stream>>>(knn_i, cnt, nbrs);
  deg_kernel<<<MM / 256, 256, 0, stream>>>(cnt, dinv);

  dim3 ggrid(MM / 128, CCc / 64);

  // ---- f_spatial = feats @ Wsp + bsp ----
  gemm_wmma_f32<<<ggrid, 256, 0, stream>>>(feats, Wsp, bsp, fsp, CCc, CCc, CCc, CCc, 0);

  // ---- Chebyshev basis (shared by low & high filters) ----
  spmm_kernel<<<MM, 256, 0, stream>>>(feats, feats, nbrs, cnt, dinv, tx1, 1.0f, 0.0f);
  spmm_kernel<<<MM, 256, 0, stream>>>(tx1, feats, nbrs, cnt, dinv, tx2, 2.0f, -1.0f);

  // ---- f_low / f_high: 3 accumulating GEMMs each ----
  gemm_wmma_f32<<<ggrid, 256, 0, stream>>>(feats, th_l,           b_low, flow, CCc, CCc, CCc, CCc, 0);
  gemm_wmma_f32<<<ggrid, 256, 0, stream>>>(tx1,   th_l + CC2,     b_low, flow, CCc, CCc, CCc, CCc, 1);
  gemm_wmma_f32<<<ggrid, 256, 0, stream>>>(tx2,   th_l + 2 * CC2, b_low, flow, CCc, CCc, CCc, CCc, 1);
  gemm_wmma_f32<<<ggrid, 256, 0, stream>>>(feats, th_h,           b_high, fhigh, CCc, CCc, CCc, CCc, 0);
  gemm_wmma_f32<<<ggrid, 256, 0, stream>>>(tx1,   th_h + CC2,     b_high, fhigh, CCc, CCc, CCc, CCc, 1);
  gemm_wmma_f32<<<ggrid, 256, 0, stream>>>(tx2,   th_h + 2 * CC2, b_high, fhigh, CCc, CCc, CCc, CCc, 1);

  // ---- h_pre = concat(fsp,flow,fhigh) @ Wg1 + bg1 (split-K over Wg1 rows) ----
  gemm_wmma_f32<<<ggrid, 256, 0, stream>>>(fsp,   Wg1,           bg1, hbuf, CCc, CCc, CCc, CCc, 0);
  gemm_wmma_f32<<<ggrid, 256, 0, stream>>>(flow,  Wg1 + CC2,     bg1, hbuf, CCc, CCc, CCc, CCc, 1);
  gemm_wmma_f32<<<ggrid, 256, 0, stream>>>(fhigh, Wg1 + 2 * CC2, bg1, hbuf, CCc, CCc, CCc, CCc, 1);

  // ---- h = relu(LN(h_pre)) in place ----
  ln_kernel<<<MM / 8, 256, 0, stream>>>(hbuf, glng, glnb, nullptr, nullptr, hbuf, 1);

  // ---- gate = softmax(h @ Wg2 + bg2) ----
  gate_kernel<<<MM / 8, 256, 0, stream>>>(hbuf, Wg2, bg2, gateb);

  // ---- f_fused (reuse tx1) ----
  float* ffuse = tx1;
  fuse_kernel<<<(int)(MC / 256), 256, 0, stream>>>(gateb, fsp, flow, fhigh, ffuse);

  // ---- o_pre = f_fused @ Wo + bo (reuse tx2) ----
  float* opre = tx2;
  gemm_wmma_f32<<<ggrid, 256, 0, stream>>>(ffuse, Wo, bo, opre, CCc, CCc, CCc, CCc, 0);

  // ---- out = feats + gamma * LN(o_pre) ----
  ln_kernel<<<MM / 8, 256, 0, stream>>>(opre, olng, olnb, feats, gamma, out, 0);
}